// SoftmaxCrossEntropyLoss_1460288881356
// MI455X (gfx1250) — compile-verified
//
#include <hip/hip_runtime.h>
#include <stdint.h>

// Softmax cross-entropy (NLL of log-softmax, mean) for [B, 6] f32 inputs,
// int64 targets. Pure HBM-bound streaming reduction:
//   268 MB / 23.3 TB/s ~= 11.5 us floor on MI455X.
// Strategy: async global->LDS (gfx1250 ASYNCcnt path) with double buffering,
// wave32 shuffle reduction, deterministic 2-pass mean.

#define TPB              256
#define C_DIM            6
#define ROWS_PER_TILE    1024
#define TILES_PER_BLOCK  4
#define ROWS_PER_BLOCK   (ROWS_PER_TILE * TILES_PER_BLOCK)      // 4096
#define TILE_FLOATS      (ROWS_PER_TILE * C_DIM)                // 6144 floats = 24 KB
#define CHUNKS_PER_TILE  (TILE_FLOATS / 4)                      // 1536 x b128
#define CHUNKS_PER_THREAD (CHUNKS_PER_TILE / TPB)               // 6
#define ROWS_PER_THREAD_T (ROWS_PER_TILE / TPB)                 // 4

typedef int   v4i __attribute__((ext_vector_type(4)));
typedef float v4f __attribute__((ext_vector_type(4)));

#if defined(__HIP_DEVICE_COMPILE__) && __has_builtin(__builtin_amdgcn_global_load_async_to_lds_b128)
#define USE_ASYNC_LDS 1
#else
#define USE_ASYNC_LDS 0
#endif

#if USE_ASYNC_LDS
#if __has_builtin(__builtin_amdgcn_s_wait_asynccnt)
#define WAIT_ASYNC(n) __builtin_amdgcn_s_wait_asynccnt(n)
#else
#define WAIT_ASYNC(n) asm volatile("s_wait_asynccnt %0" ::"i"(n) : "memory")
#endif
#else
#define WAIT_ASYNC(n) ((void)0)
#endif

__device__ __forceinline__ void issue_tile_load(const float* __restrict__ gbase,
                                                float* __restrict__ lbase, int tid) {
#if USE_ASYNC_LDS
#pragma unroll
  for (int j = 0; j < CHUNKS_PER_THREAD; ++j) {
    const int c = tid + j * TPB;
    // async b128: global -> LDS, tracked by ASYNCcnt (gfx1250).
    // Generic pointers with the builtin's expected pointee type (int4);
    // clang inserts the generic->global / generic->LDS addrspace casts.
    __builtin_amdgcn_global_load_async_to_lds_b128(
        (v4i*)(gbase + (size_t)c * 4), (v4i*)(lbase + c * 4),
        /*offset=*/0, /*cpol=*/0);
  }
#else
#pragma unroll
  for (int j = 0; j < CHUNKS_PER_THREAD; ++j) {
    const int c = tid + j * TPB;
    v4f v = __builtin_nontemporal_load((const v4f*)gbase + c);
    *((v4f*)lbase + c) = v;
  }
#endif
}

__device__ __forceinline__ float row_nll(float v0, float v1, float v2, float v3,
                                         float v4, float v5, int t) {
  const float m = fmaxf(fmaxf(fmaxf(v0, v1), fmaxf(v2, v3)), fmaxf(v4, v5));
  const float s = __expf(v0 - m) + __expf(v1 - m) + __expf(v2 - m) +
                  __expf(v3 - m) + __expf(v4 - m) + __expf(v5 - m);
  float xt = v0;
  xt = (t == 1) ? v1 : xt;
  xt = (t == 2) ? v2 : xt;
  xt = (t == 3) ? v3 : xt;
  xt = (t == 4) ? v4 : xt;
  xt = (t == 5) ? v5 : xt;
  return m + __logf(s) - xt;
}

__global__ __launch_bounds__(TPB) void ce_partial_kernel(
    const float* __restrict__ x, const long long* __restrict__ tgt,
    float* __restrict__ partial, int rows) {
  __shared__ alignas(16) float buf[2][TILE_FLOATS];
  __shared__ float wsum[TPB / 32];

  const int tid = threadIdx.x;
  const int blockRowBase = blockIdx.x * ROWS_PER_BLOCK;
  float acc = 0.0f;

  if (blockRowBase + ROWS_PER_BLOCK <= rows) {
    // ---- fast path: full tiles, async double-buffered LDS staging ----
    issue_tile_load(x + (size_t)blockRowBase * C_DIM, buf[0], tid);
#pragma unroll
    for (int t = 0; t < TILES_PER_BLOCK; ++t) {
      const int cur = t & 1;
      const int tileRowBase = blockRowBase + t * ROWS_PER_TILE;
      if (t + 1 < TILES_PER_BLOCK) {
        issue_tile_load(x + (size_t)(tileRowBase + ROWS_PER_TILE) * C_DIM,
                        buf[cur ^ 1], tid);
        WAIT_ASYNC(CHUNKS_PER_THREAD);  // current tile's 6 ops done (in-order)
      } else {
        WAIT_ASYNC(0);
      }
      __syncthreads();  // all waves' async writes to buf[cur] visible
#pragma unroll
      for (int r = 0; r < ROWS_PER_THREAD_T; ++r) {
        const int lrow = tid * ROWS_PER_THREAD_T + r;
        const float* rowp = &buf[cur][lrow * C_DIM];
        const long long tg =
            __builtin_nontemporal_load(&tgt[(size_t)tileRowBase + lrow]);
        acc += row_nll(rowp[0], rowp[1], rowp[2], rowp[3], rowp[4], rowp[5],
                       (int)tg);
      }
      __syncthreads();  // buf[cur] free for the t+2 prefetch
    }
  } else {
    // ---- tail path: guarded direct global loads ----
    const int rowEnd =
        (rows < blockRowBase + ROWS_PER_BLOCK) ? rows : (blockRowBase + ROWS_PER_BLOCK);
    for (int row = blockRowBase + tid; row < rowEnd; row += TPB) {
      const float* rowp = x + (size_t)row * C_DIM;
      const long long tg = tgt[row];
      acc += row_nll(rowp[0], rowp[1], rowp[2], rowp[3], rowp[4], rowp[5],
                     (int)tg);
    }
  }

  // ---- wave32 + cross-wave reduction ----
#pragma unroll
  for (int off = 16; off > 0; off >>= 1) acc += __shfl_down(acc, off, 32);
  if ((tid & 31) == 0) wsum[tid >> 5] = acc;
  __syncthreads();
  if (tid == 0) {
    float s = 0.0f;
#pragma unroll
    for (int w = 0; w < TPB / 32; ++w) s += wsum[w];
    partial[blockIdx.x] = s;
  }
}

__global__ __launch_bounds__(TPB) void ce_final_kernel(
    const float* __restrict__ partial, int n, float* __restrict__ out,
    float scale) {
  __shared__ float wsum[TPB / 32];
  const int tid = threadIdx.x;
  float acc = 0.0f;
  for (int i = tid; i < n; i += TPB) acc += partial[i];
#pragma unroll
  for (int off = 16; off > 0; off >>= 1) acc += __shfl_down(acc, off, 32);
  if ((tid & 31) == 0) wsum[tid >> 5] = acc;
  __syncthreads();
  if (tid == 0) {
    float s = 0.0f;
#pragma unroll
    for (int w = 0; w < TPB / 32; ++w) s += wsum[w];
    out[0] = s * scale;
  }
}

extern "C" void kernel_launch(void* const* d_in, const int* in_sizes, int n_in,
                              void* d_out, int out_size, void* d_ws,
                              size_t ws_size, hipStream_t stream) {
  (void)n_in;
  (void)out_size;
  (void)ws_size;
  const float* x = (const float*)d_in[0];
  const long long* tgt = (const long long*)d_in[1];
  float* out = (float*)d_out;
  const int rows = in_sizes[1];  // B (targets count)
  const int blocks = (rows + ROWS_PER_BLOCK - 1) / ROWS_PER_BLOCK;  // 2048 for B=8.39M
  float* partial = (float*)d_ws;  // blocks * 4 bytes

  ce_partial_kernel<<<blocks, TPB, 0, stream>>>(x, tgt, partial, rows);
  ce_final_kernel<<<1, TPB, 0, stream>>>(partial, blocks, out,
                                         1.0f / (float)rows);
}